// Estimate_Covariance_24352464569636
// MI455X (gfx1250) — compile-verified
//
#include <hip/hip_runtime.h>
#include <hip/hip_bf16.h>
#include <stdint.h>

#define C_CLS 1000
#define A_DIM 512
#define N_ROWS 65536
#define MOM 0.8f
#define CAP 2048   // max rows/class held in LDS (expected ~65, binomial tail << 2048)

#if defined(__HIP_DEVICE_COMPILE__) && defined(__gfx1250__) && \
    __has_builtin(__builtin_amdgcn_global_load_async_to_lds_b128) && \
    __has_builtin(__builtin_amdgcn_global_load_async_to_lds_b32) && \
    __has_builtin(__builtin_amdgcn_s_wait_asynccnt)
#define HAVE_ASYNC_LDS 1
#else
#define HAVE_ASYNC_LDS 0
#endif

// ---------------- K4 (defined first so disasm snippet shows it):
// per-class reduce + finalize ----------------
__global__ __launch_bounds__(256) void k_reduce(
    const float* __restrict__ feat, const float* __restrict__ cov,
    const float* __restrict__ mean, const float* __restrict__ amount,
    const unsigned* __restrict__ cnt, const unsigned* __restrict__ base,
    const unsigned* __restrict__ rowidx,
    float* __restrict__ out_cov, float* __restrict__ out_mean,
    float* __restrict__ out_amt) {
    const int c = blockIdx.x;
    const int t = threadIdx.x;

    __shared__ unsigned rows_s[CAP];           // 8 KB
    __shared__ float buf[2][4 * A_DIM];        // 16 KB: 2 buffers x 4 rows x 512 f32

    const unsigned n = cnt[c];
    const unsigned b0 = base[c];
    const int m = (int)(n < (unsigned)CAP ? n : (unsigned)CAP);

    // Stage the class's row list into LDS.
#if HAVE_ASYNC_LDS
    {
        typedef __attribute__((address_space(1))) int* gi32_p;
        typedef __attribute__((address_space(3))) int* li32_p;
        gi32_p gidx = (gi32_p)(rowidx + b0);
        li32_p lidx = (li32_p)(&rows_s[0]);
        for (int i = t; i < m; i += 256)
            __builtin_amdgcn_global_load_async_to_lds_b32(gidx + i, lidx + i, 0, 0);
        __builtin_amdgcn_s_wait_asynccnt(0);
    }
#else
    for (int i = t; i < m; i += 256) rows_s[i] = rowidx[b0 + i];
#endif
    __syncthreads();

    // Odd-even transposition sort of the row list -> deterministic f32 sum order.
    for (int ph = 0; ph < m; ++ph) {
        for (int i = (ph & 1) + 2 * t; i + 1 < m; i += 512) {
            unsigned a = rows_s[i], b = rows_s[i + 1];
            if (a > b) { rows_s[i] = b; rows_s[i + 1] = a; }
        }
        __syncthreads();
    }

    float s0 = 0.f, s1 = 0.f, q0 = 0.f, q1 = 0.f;
    const int col = t * 2;   // this thread owns columns col, col+1

#if HAVE_ASYNC_LDS
    typedef int v4i __attribute__((ext_vector_type(4)));
    typedef __attribute__((address_space(1))) v4i* gv4i_p;
    typedef __attribute__((address_space(3))) v4i* lv4i_p;
    gv4i_p gbase = (gv4i_p)feat;                 // global feature base, 16B units
    lv4i_p lbase = (lv4i_p)(&buf[0][0]);         // LDS buffer base, 16B units

    const int unit = t & 127;   // 16-byte unit within a 2 KB row
    const int half = t >> 7;    // which of the 2 rows this half-block copies
    const int nch = (m + 3) >> 2;

    auto issue = [&](int ch, int pb) {
#pragma unroll
        for (int i = 0; i < 2; ++i) {
            int r = ch * 4 + i * 2 + half;
            int rr = (r < m) ? r : (m - 1);          // clamp: keeps issue count uniform
            unsigned row = rows_s[rr];
            __builtin_amdgcn_global_load_async_to_lds_b128(
                gbase + (size_t)row * (A_DIM / 4) + unit,
                lbase + pb * (4 * A_DIM / 4) + (i * 2 + half) * (A_DIM / 4) + unit,
                0, 0);
        }
    };

    if (nch > 0) issue(0, 0);
    for (int ch = 0; ch < nch; ++ch) {
        const int nxt = ch + 1;
        if (nxt < nch) {
            issue(nxt, nxt & 1);                       // prefetch next chunk
            __builtin_amdgcn_s_wait_asynccnt(2);       // retire current chunk only
        } else {
            __builtin_amdgcn_s_wait_asynccnt(0);
        }
        __syncthreads();
        const float* B = buf[ch & 1];
#pragma unroll
        for (int r = 0; r < 4; ++r) {
            int g = ch * 4 + r;
            if (g < m) {
                float2 v = *(const float2*)&B[r * A_DIM + col];   // ds_load_b64
                s0 += v.x; q0 = fmaf(v.x, v.x, q0);
                s1 += v.y; q1 = fmaf(v.y, v.y, q1);
            }
        }
        __syncthreads();
    }
#else
    for (int i = 0; i < m; ++i) {
        unsigned row = rows_s[i];
        float2 v = *(const float2*)(feat + (size_t)row * A_DIM + col);
        s0 += v.x; q0 = fmaf(v.x, v.x, q0);
        s1 += v.y; q1 = fmaf(v.y, v.y, q1);
    }
#endif

    // Pathological overflow tail (never taken for this input distribution).
    for (unsigned i = (unsigned)CAP; i < n; ++i) {
        unsigned row = rowidx[b0 + i];
        float2 v = *(const float2*)(feat + (size_t)row * A_DIM + col);
        s0 += v.x; q0 = fmaf(v.x, v.x, q0);
        s1 += v.y; q1 = fmaf(v.y, v.y, q1);
    }

    // Finalize: w, blend cov/mean, write amount.
    const float cntf = (float)n;
    const float safe = (n == 0u) ? 1.0f : cntf;
    const float amt = amount[c];
    const float den = cntf + amt;
    const float w_raw = (den > 0.f) ? (cntf / den) : 0.f;
    const float w = (w_raw > 0.f) ? fmaxf(w_raw, 1.0f - MOM) : 0.f;
    const float omw = 1.0f - w;

    const size_t o = (size_t)c * A_DIM + col;
    float ss[2] = { s0, s1 }, qq[2] = { q0, q1 };
#pragma unroll
    for (int k = 0; k < 2; ++k) {
        float ave = ss[k] / safe;
        float var = fmaxf(qq[k] / safe - ave * ave, 0.0f);
        float mv = mean[o + k];
        float cv = cov[o + k];
        float d = mv - ave;
        float add = w * omw * d * d;
        out_cov[o + k] = cv * omw + var * w + add;
        out_mean[o + k] = mv * omw + ave * w;
    }
    if (t == 0) out_amt[c] = amt + cntf;
}

// ---------------- K0: zero class counters ----------------
__global__ void k_zero(unsigned* __restrict__ cnt) {
    int i = blockIdx.x * 256 + threadIdx.x;
    if (i < C_CLS) cnt[i] = 0u;
}

// ---------------- K1: per-class counts ----------------
__global__ void k_count(const int* __restrict__ labels, unsigned* __restrict__ cnt) {
    int n = blockIdx.x * 256 + threadIdx.x;
    if (n < N_ROWS) {
        int c = labels[n];
        atomicAdd(&cnt[c], 1u);
    }
}

// ---------------- K2: exclusive scan over 1000 counts (single block) ----------------
__global__ __launch_bounds__(1024) void k_scan(const unsigned* __restrict__ cnt,
                                               unsigned* __restrict__ base,
                                               unsigned* __restrict__ cursor) {
    __shared__ unsigned sd[1024];
    int t = threadIdx.x;
    unsigned v = (t < C_CLS) ? cnt[t] : 0u;
    sd[t] = v;
    __syncthreads();
    for (int off = 1; off < 1024; off <<= 1) {
        unsigned x = (t >= off) ? sd[t - off] : 0u;
        __syncthreads();
        sd[t] += x;
        __syncthreads();
    }
    if (t < C_CLS) {
        unsigned e = sd[t] - v;   // exclusive prefix
        base[t] = e;
        cursor[t] = e;
    }
}

// ---------------- K3: bin row indices per class ----------------
__global__ void k_scatter(const int* __restrict__ labels,
                          unsigned* __restrict__ cursor,
                          unsigned* __restrict__ rowidx) {
    int n = blockIdx.x * 256 + threadIdx.x;
    if (n < N_ROWS) {
        int c = labels[n];
        unsigned p = atomicAdd(&cursor[c], 1u);
        rowidx[p] = (unsigned)n;   // order fixed later by in-LDS sort
    }
}

extern "C" void kernel_launch(void* const* d_in, const int* in_sizes, int n_in,
                              void* d_out, int out_size, void* d_ws, size_t ws_size,
                              hipStream_t stream) {
    (void)in_sizes; (void)n_in; (void)out_size; (void)ws_size;
    const float* feat   = (const float*)d_in[0];
    const int*   labels = (const int*)d_in[1];
    const float* cov    = (const float*)d_in[2];
    const float* mean   = (const float*)d_in[3];
    const float* amount = (const float*)d_in[4];

    float* out_cov  = (float*)d_out;
    float* out_mean = out_cov + (size_t)C_CLS * A_DIM;
    float* out_amt  = out_mean + (size_t)C_CLS * A_DIM;

    unsigned* cnt    = (unsigned*)d_ws;       // 1024 u32
    unsigned* base   = cnt + 1024;            // 1024 u32
    unsigned* cursor = base + 1024;           // 1024 u32
    unsigned* rowidx = cursor + 1024;         // 65536 u32  (total ws: ~268 KB)

    k_zero<<<(C_CLS + 255) / 256, 256, 0, stream>>>(cnt);
    k_count<<<N_ROWS / 256, 256, 0, stream>>>(labels, cnt);
    k_scan<<<1, 1024, 0, stream>>>(cnt, base, cursor);
    k_scatter<<<N_ROWS / 256, 256, 0, stream>>>(labels, cursor, rowidx);
    k_reduce<<<C_CLS, 256, 0, stream>>>(feat, cov, mean, amount, cnt, base, rowidx,
                                        out_cov, out_mean, out_amt);
}